// MultiHeadAttentionBlock_9371618640030
// MI455X (gfx1250) — compile-verified
//
#include <hip/hip_runtime.h>
#include <math.h>

// ---------------------------------------------------------------------------
// Multi-head attention block for MI455X (gfx1250), exact-fp32 WMMA path.
//   D_MODEL=1024, H=16, d_k=64, B=2, S=2048  ->  4096 tokens.
// All matmuls use V_WMMA_F32_16X16X4_F32 (exact fp32, K=4 per instruction).
// ---------------------------------------------------------------------------

typedef float v2f __attribute__((ext_vector_type(2)));
typedef float v8f __attribute__((ext_vector_type(8)));

#define D_MODEL 1024
#define NHEAD   16
#define DK      64
#define BATCH   2
#define SEQ     2048
#define NTOK    (BATCH * SEQ)   // 4096

__device__ __forceinline__ v8f wmma_f32x4(v2f a, v2f b, v8f c) {
  // D = A(16x4 f32) * B(4x16 f32) + C(16x16 f32)
  return __builtin_amdgcn_wmma_f32_16x16x4_f32(
      /*neg_a=*/false, a, /*neg_b=*/false, b,
      /*c_mod=*/(short)0, c, /*reuse_a=*/false, /*reuse_b=*/false);
}

// ---------------------------------------------------------------------------
// C[M,N] = A[M,K] * W[N,K]^T + bias[N]       (row-major everything, "NT" GEMM)
// Block: 256 threads = 8 waves in a 4(m) x 2(n) grid; each wave computes a
// 32x64 tile (2 m-subtiles x 4 n-subtiles = 8 fp32 WMMA accumulators), so a
// workgroup covers a 128x128 tile of C. Per k-step a wave issues 6 b64 loads
// and 8 WMMAs -> ~10.7 FLOP/B from L2, with 8 independent WMMAs in flight.
// A-frag lane addressing == B-frag lane addressing (both row-major, K inner):
//   lane L loads float2 at row[L%16]*K + k + (L/16)*2
// ---------------------------------------------------------------------------
__global__ __launch_bounds__(256)
void gemm_nt_bias(const float* __restrict__ A, const float* __restrict__ W,
                  const float* __restrict__ bias, float* __restrict__ C,
                  int M, int N, int K)
{
  const int lane = threadIdx.x & 31;
  const int wave = threadIdx.x >> 5;
  const int half = lane >> 4;    // 0 or 1
  const int l16  = lane & 15;

  const int mblk = blockIdx.y * 128 + (wave >> 1) * 32;   // wave m-origin
  const int nblk = blockIdx.x * 128 + (wave & 1) * 64;    // wave n-origin

  const float* arow0 = A + (size_t)(mblk +      l16) * K + half * 2;
  const float* arow1 = A + (size_t)(mblk + 16 + l16) * K + half * 2;
  const float* wrow0 = W + (size_t)(nblk +      l16) * K + half * 2;
  const float* wrow1 = W + (size_t)(nblk + 16 + l16) * K + half * 2;
  const float* wrow2 = W + (size_t)(nblk + 32 + l16) * K + half * 2;
  const float* wrow3 = W + (size_t)(nblk + 48 + l16) * K + half * 2;

  v8f acc[2][4];
  #pragma unroll
  for (int i = 0; i < 2; ++i)
    #pragma unroll
    for (int j = 0; j < 4; ++j) acc[i][j] = (v8f){};

  for (int k = 0; k < K; k += 4) {
    v2f a0 = *(const v2f*)(arow0 + k);
    v2f a1 = *(const v2f*)(arow1 + k);
    v2f b0 = *(const v2f*)(wrow0 + k);
    v2f b1 = *(const v2f*)(wrow1 + k);
    v2f b2 = *(const v2f*)(wrow2 + k);
    v2f b3 = *(const v2f*)(wrow3 + k);
    acc[0][0] = wmma_f32x4(a0, b0, acc[0][0]);
    acc[0][1] = wmma_f32x4(a0, b1, acc[0][1]);
    acc[0][2] = wmma_f32x4(a0, b2, acc[0][2]);
    acc[0][3] = wmma_f32x4(a0, b3, acc[0][3]);
    acc[1][0] = wmma_f32x4(a1, b0, acc[1][0]);
    acc[1][1] = wmma_f32x4(a1, b1, acc[1][1]);
    acc[1][2] = wmma_f32x4(a1, b2, acc[1][2]);
    acc[1][3] = wmma_f32x4(a1, b3, acc[1][3]);
  }

  // Epilogue: C layout -> lane holds rows m=(lane/16)*8+r, col n=lane%16.
  #pragma unroll
  for (int j = 0; j < 4; ++j) {
    const float bb = bias[nblk + j * 16 + l16];
    #pragma unroll
    for (int i = 0; i < 2; ++i) {
      #pragma unroll
      for (int r = 0; r < 8; ++r) {
        const int m = mblk + i * 16 + half * 8 + r;
        C[(size_t)m * N + nblk + j * 16 + l16] = acc[i][j][r] + bb;
      }
    }
  }
}

// ---------------------------------------------------------------------------
// Causal flash attention over projected Q/K/V stored as [NTOK, D_MODEL]
// (head h occupies columns h*64 .. h*64+63, matching split_heads).
// Grid: (S/64, B*H). Block: 128 threads = 4 waves; each wave owns one
// 16-row query tile; streams 16-wide KV blocks up to the causal diagonal
// with online softmax. Output written as ctx[b, q, h*64+d] (merged heads),
// so the out-projection is just another NT GEMM.
// ---------------------------------------------------------------------------
__global__ __launch_bounds__(128)
void attention_causal(const float* __restrict__ Qp, const float* __restrict__ Kp,
                      const float* __restrict__ Vp, float* __restrict__ Ctx)
{
  __shared__ float pbuf[4][16][17];   // per-wave P tile, padded vs bank conflicts

  const int lane = threadIdx.x & 31;
  const int wave = threadIdx.x >> 5;
  const int half = lane >> 4;
  const int l16  = lane & 15;

  const int bh   = blockIdx.y;
  const int b    = bh / NHEAD;
  const int h    = bh % NHEAD;
  const size_t rowbase = (size_t)b * SEQ;     // token-row offset for this batch
  const int hoff  = h * DK;
  const int qbase = blockIdx.x * 64 + wave * 16;

  // Preload Q fragments for the whole d_k=64: 16 k-steps of float2 per lane.
  v2f qf[16];
  {
    const float* qrow = Qp + (rowbase + qbase + l16) * (size_t)D_MODEL + hoff + half * 2;
    #pragma unroll
    for (int i = 0; i < 16; ++i) qf[i] = *(const v2f*)(qrow + i * 4);
  }

  float m_i[8], l_i[8];
  v8f o0 = {}, o1 = {}, o2 = {}, o3 = {};
  #pragma unroll
  for (int r = 0; r < 8; ++r) { m_i[r] = -1e30f; l_i[r] = 0.0f; }

  const int nblocks = qbase / 16 + 1;   // causal: only KV blocks <= diagonal
  for (int jb = 0; jb < nblocks; ++jb) {
    const int jbase = jb * 16;

    // ---- S = Q * K^T (16x16 tile, 16 fp32 WMMAs over d_k=64) ----
    v8f s = {};
    const float* krow = Kp + (rowbase + jbase + l16) * (size_t)D_MODEL + hoff + half * 2;
    #pragma unroll
    for (int i = 0; i < 16; ++i) {
      v2f kb = *(const v2f*)(krow + i * 4);
      s = wmma_f32x4(qf[i], kb, s);
    }

    // ---- scale, causal mask, online softmax update ----
    float corr[8];
    #pragma unroll
    for (int r = 0; r < 8; ++r) {
      float sv = s[r] * 0.125f;                    // 1/sqrt(64)
      const int qidx = qbase + half * 8 + r;
      const int kidx = jbase + l16;
      if (kidx > qidx) sv = -1e30f;

      float mx = sv;
      mx = fmaxf(mx, __shfl_xor(mx, 1, 32));
      mx = fmaxf(mx, __shfl_xor(mx, 2, 32));
      mx = fmaxf(mx, __shfl_xor(mx, 4, 32));
      mx = fmaxf(mx, __shfl_xor(mx, 8, 32));       // row max within 16-lane half

      const float mnew = fmaxf(m_i[r], mx);
      const float c    = __expf(m_i[r] - mnew);
      const float pv   = __expf(sv - mnew);

      float rs = pv;
      rs += __shfl_xor(rs, 1, 32);
      rs += __shfl_xor(rs, 2, 32);
      rs += __shfl_xor(rs, 4, 32);
      rs += __shfl_xor(rs, 8, 32);                 // row sum

      l_i[r] = l_i[r] * c + rs;
      m_i[r] = mnew;
      corr[r] = c;
      pbuf[wave][half * 8 + r][l16] = pv;          // C-layout -> LDS (row-major)
    }

    // rescale running output
    #pragma unroll
    for (int r = 0; r < 8; ++r) {
      o0[r] *= corr[r]; o1[r] *= corr[r]; o2[r] *= corr[r]; o3[r] *= corr[r];
    }

    // ---- O += P * V  (P re-read from LDS in A-layout; intra-wave DS is
    // in-order so no barrier needed). 4 n-tiles x 4 k-steps = 16 WMMAs. ----
    const float* vcol = Vp + (rowbase + jbase) * (size_t)D_MODEL + hoff + l16;
    #pragma unroll
    for (int k0 = 0; k0 < 16; k0 += 4) {
      const int kk = k0 + half * 2;
      v2f a;
      a.x = pbuf[wave][l16][kk];
      a.y = pbuf[wave][l16][kk + 1];
      const float* v0 = vcol + (size_t)kk * D_MODEL;
      const float* v1 = vcol + (size_t)(kk + 1) * D_MODEL;
      v2f bv;
      bv.x = v0[ 0]; bv.y = v1[ 0]; o0 = wmma_f32x4(a, bv, o0);
      bv.x = v0[16]; bv.y = v1[16]; o1 = wmma_f32x4(a, bv, o1);
      bv.x = v0[32]; bv.y = v1[32]; o2 = wmma_f32x4(a, bv, o2);
      bv.x = v0[48]; bv.y = v1[48]; o3 = wmma_f32x4(a, bv, o3);
    }
  }

  // ---- normalize and write ctx[b, q, h*64 + d] ----
  #pragma unroll
  for (int r = 0; r < 8; ++r) {
    const float inv = 1.0f / l_i[r];
    const size_t row = (rowbase + qbase + half * 8 + r) * (size_t)D_MODEL + hoff;
    Ctx[row +      l16] = o0[r] * inv;
    Ctx[row + 16 + l16] = o1[r] * inv;
    Ctx[row + 32 + l16] = o2[r] * inv;
    Ctx[row + 48 + l16] = o3[r] * inv;
  }
}

// ---------------------------------------------------------------------------
// Launch: 3 projection GEMMs -> causal attention -> output GEMM.
// d_ws holds Qp/Kp/Vp/Ctx (4 x 16 MB fp32), all fully overwritten each call.
// ---------------------------------------------------------------------------
extern "C" void kernel_launch(void* const* d_in, const int* in_sizes, int n_in,
                              void* d_out, int out_size, void* d_ws, size_t ws_size,
                              hipStream_t stream) {
  const float* query  = (const float*)d_in[0];
  const float* key_in = (const float*)d_in[1];
  const float* value  = (const float*)d_in[2];
  // d_in[3] is the causal mask (int32) -- causality is computed analytically.
  const float* w_q   = (const float*)d_in[4];
  const float* b_q   = (const float*)d_in[5];
  const float* w_k   = (const float*)d_in[6];
  const float* b_k   = (const float*)d_in[7];
  const float* w_v   = (const float*)d_in[8];
  const float* b_v   = (const float*)d_in[9];
  const float* w_out = (const float*)d_in[10];
  const float* b_out = (const float*)d_in[11];
  float* out = (float*)d_out;

  const size_t mat = (size_t)NTOK * D_MODEL;   // 4096*1024 floats
  float* Qp  = (float*)d_ws;
  float* Kp  = Qp + mat;
  float* Vp  = Kp + mat;
  float* Ctx = Vp + mat;

  const dim3 gblk(256);
  const dim3 ggrid(D_MODEL / 128, NTOK / 128);   // (8, 32)

  gemm_nt_bias<<<ggrid, gblk, 0, stream>>>(query,  w_q, b_q, Qp, NTOK, D_MODEL, D_MODEL);
  gemm_nt_bias<<<ggrid, gblk, 0, stream>>>(key_in, w_k, b_k, Kp, NTOK, D_MODEL, D_MODEL);
  gemm_nt_bias<<<ggrid, gblk, 0, stream>>>(value,  w_v, b_v, Vp, NTOK, D_MODEL, D_MODEL);

  const dim3 agrid(SEQ / 64, BATCH * NHEAD);     // (32, 32)
  attention_causal<<<agrid, dim3(128), 0, stream>>>(Qp, Kp, Vp, Ctx);

  gemm_nt_bias<<<ggrid, gblk, 0, stream>>>(Ctx, w_out, b_out, out, NTOK, D_MODEL, D_MODEL);
}